// Aggregator1__26886495273089
// MI455X (gfx1250) — compile-verified
//
#include <hip/hip_runtime.h>

typedef __attribute__((ext_vector_type(2))) float v2f;
typedef __attribute__((ext_vector_type(8))) float v8f;

#define DIM 128
#define LDR 132   // LDS row pitch: 132 % 64 = 4 -> conflict-free across 16 rows

__device__ __forceinline__ void atomic_add_f32g(float* p, float v) {
  // native no-return global atomic add (L2 atomic unit); respects EXEC
  asm volatile("global_atomic_add_f32 %0, %1, off" : : "v"(p), "v"(v) : "memory");
}

__device__ __forceinline__ v8f wmma_f32(v2f a, v2f b, v8f c) {
  return __builtin_amdgcn_wmma_f32_16x16x4_f32(false, a, false, b, (short)0, c,
                                               false, false);
}

// Edge aggregation:
//   DUAL=1 (t-path): rows = (A1@Wa^T)*(V1@Wv^T) + (A2@Wa^T)*(V2@Wv^T), *scale
//   DUAL=0 (v-path): rows = (A1@Wa^T)*(V1@Wv^T) * scale
// CSR scatter-add into dst with per-lane run-length compression of equal
// segment ids (CSR edges are sorted -> ~8 edges/segment).
// Block = 256 threads (8 wave32); one 16-edge M-tile/block, 16-col N-tile/wave.
template <bool DUAL>
__launch_bounds__(256)
__global__ void edge_agg_wmma(const float* __restrict__ A1src,
                              const int* __restrict__ A1idx,
                              const float* __restrict__ V1src,
                              const int* __restrict__ V1idx,
                              const float* __restrict__ A2src,  // direct (E,128)
                              const float* __restrict__ V2src,  // direct (E,128)
                              const float* __restrict__ Wa,     // (128,128) as W^T
                              const float* __restrict__ Wv,     // (128,128) as W^T
                              const int* __restrict__ ptr,      // (nseg+1) CSR
                              int nseg, float scale,
                              float* __restrict__ dst)          // (nseg,128)
{
  extern __shared__ float smem[];
  float* lds_a1 = smem;
  float* lds_v1 = smem + 16 * LDR;
  float* lds_a2 = smem + 2 * 16 * LDR;  // used only when DUAL
  float* lds_v2 = smem + 3 * 16 * LDR;
  int*   lds_seg = (int*)(smem + (DUAL ? 4 : 2) * 16 * LDR);

  const int tid = threadIdx.x;
  const int m0  = blockIdx.x * 16;

  // Stage the 16-row operand tiles into LDS (float4 chunks -> ds_store_b128).
  for (int i = tid; i < 16 * 32; i += 256) {
    const int row = i >> 5;
    const int c4  = (i & 31) << 2;
    const float* ap = A1idx ? (A1src + (size_t)A1idx[m0 + row] * DIM)
                            : (A1src + (size_t)(m0 + row) * DIM);
    const float* vp = V1idx ? (V1src + (size_t)V1idx[m0 + row] * DIM)
                            : (V1src + (size_t)(m0 + row) * DIM);
    *(float4*)&lds_a1[row * LDR + c4] = *(const float4*)&ap[c4];
    *(float4*)&lds_v1[row * LDR + c4] = *(const float4*)&vp[c4];
    if (DUAL) {
      *(float4*)&lds_a2[row * LDR + c4] =
          *(const float4*)&A2src[(size_t)(m0 + row) * DIM + c4];
      *(float4*)&lds_v2[row * LDR + c4] =
          *(const float4*)&V2src[(size_t)(m0 + row) * DIM + c4];
    }
  }
  // CSR segment id: seg(e) = searchsorted(ptr, e, 'right') - 1
  if (tid < 16) {
    const int e = m0 + tid;
    int lo = 0, hi = nseg + 1;
    while (lo < hi) {
      const int mid = (lo + hi) >> 1;
      if (ptr[mid] <= e) lo = mid + 1; else hi = mid;
    }
    int s = lo - 1;
    s = s < 0 ? 0 : (s >= nseg ? nseg - 1 : s);
    lds_seg[tid] = s;
  }
  __syncthreads();

  const int wave = tid >> 5;
  const int lane = tid & 31;
  const int col  = lane & 15;   // A: M row; B/C/D: N column
  const int h    = lane >> 4;   // lane-half -> K offset / M-half
  const int n0   = wave * 16;

  const float* wa_row = Wa + (size_t)(n0 + col) * DIM + 2 * h;  // B[k][n]=Wa[n][k]
  const float* wv_row = Wv + (size_t)(n0 + col) * DIM + 2 * h;
  const float* la1 = &lds_a1[col * LDR + 2 * h];
  const float* lv1 = &lds_v1[col * LDR + 2 * h];
  const float* la2 = &lds_a2[col * LDR + 2 * h];
  const float* lv2 = &lds_v2[col * LDR + 2 * h];

  v8f p1 = {}, q1 = {}, p2 = {}, q2 = {};
#pragma unroll 4
  for (int k0 = 0; k0 < DIM; k0 += 4) {
    const v2f ba = *(const v2f*)&wa_row[k0];
    const v2f bv = *(const v2f*)&wv_row[k0];
    p1 = wmma_f32(*(const v2f*)&la1[k0], ba, p1);
    q1 = wmma_f32(*(const v2f*)&lv1[k0], bv, q1);
    if (DUAL) {
      p2 = wmma_f32(*(const v2f*)&la2[k0], ba, p2);
      q2 = wmma_f32(*(const v2f*)&lv2[k0], bv, q2);
    }
  }

  // Gather this lane-half's 8 segment ids in two b128 LDS loads.
  int segs[8];
  {
    const int4 s0 = *(const int4*)&lds_seg[8 * h];
    const int4 s1 = *(const int4*)&lds_seg[8 * h + 4];
    segs[0] = s0.x; segs[1] = s0.y; segs[2] = s0.z; segs[3] = s0.w;
    segs[4] = s1.x; segs[5] = s1.y; segs[6] = s1.z; segs[7] = s1.w;
  }
  float d[8];
#pragma unroll
  for (int i = 0; i < 8; ++i) {
    float v = p1[i] * q1[i];
    if (DUAL) v += p2[i] * q2[i];
    d[i] = v * scale;
  }

  // Run-length compressed scatter: one atomic per distinct segment run.
  float* dstb = dst + n0 + col;
  float run = d[0];
  int   cur = segs[0];
#pragma unroll
  for (int i = 1; i < 8; ++i) {
    if (segs[i] == cur) {
      run += d[i];
    } else {
      atomic_add_f32g(dstb + (size_t)cur * DIM, run);
      cur = segs[i];
      run = d[i];
    }
  }
  atomic_add_f32g(dstb + (size_t)cur * DIM, run);
}

// out[m][n] = sum_k X1[m][k]*W[n][k] + sum_k X2[m][k]*W[n][k+128],  W (128,256)
__launch_bounds__(256)
__global__ void final_concat_gemm(const float* __restrict__ X1,
                                  const float* __restrict__ X2,
                                  const float* __restrict__ W,
                                  float* __restrict__ out)
{
  const int tid  = threadIdx.x;
  const int m0   = blockIdx.x * 16;
  const int wave = tid >> 5;
  const int lane = tid & 31;
  const int col  = lane & 15;
  const int h    = lane >> 4;
  const int n0   = wave * 16;

  const float* wrow = W + (size_t)(n0 + col) * (2 * DIM) + 2 * h;
  const float* x1   = X1 + (size_t)(m0 + col) * DIM + 2 * h;
  const float* x2   = X2 + (size_t)(m0 + col) * DIM + 2 * h;

  v8f acc = {};
#pragma unroll 4
  for (int k0 = 0; k0 < DIM; k0 += 4) {
    acc = wmma_f32(*(const v2f*)&x1[k0], *(const v2f*)&wrow[k0], acc);
  }
#pragma unroll 4
  for (int k0 = 0; k0 < DIM; k0 += 4) {
    acc = wmma_f32(*(const v2f*)&x2[k0], *(const v2f*)&wrow[DIM + k0], acc);
  }
#pragma unroll
  for (int i = 0; i < 8; ++i)
    out[(size_t)(m0 + i + 8 * h) * DIM + n0 + col] = acc[i];
}

// out = X @ W with W (128,128) row-major NOT transposed: B[k][n] = W[k][n]
__launch_bounds__(256)
__global__ void final_gemm_nn(const float* __restrict__ X,
                              const float* __restrict__ W,
                              float* __restrict__ out)
{
  const int tid  = threadIdx.x;
  const int m0   = blockIdx.x * 16;
  const int wave = tid >> 5;
  const int lane = tid & 31;
  const int col  = lane & 15;
  const int h    = lane >> 4;
  const int n0   = wave * 16;

  const float* x = X + (size_t)(m0 + col) * DIM + 2 * h;

  v8f acc = {};
#pragma unroll 4
  for (int k0 = 0; k0 < DIM; k0 += 4) {
    v2f b;
    b.x = W[(size_t)(k0 + 2 * h) * DIM + n0 + col];
    b.y = W[(size_t)(k0 + 2 * h + 1) * DIM + n0 + col];
    acc = wmma_f32(*(const v2f*)&x[k0], b, acc);
  }
#pragma unroll
  for (int i = 0; i < 8; ++i)
    out[(size_t)(m0 + i + 8 * h) * DIM + n0 + col] = acc[i];
}

__global__ void zero_ws(float4* __restrict__ p, int n4) {
  const int i = blockIdx.x * blockDim.x + threadIdx.x;
  if (i < n4) p[i] = make_float4(0.f, 0.f, 0.f, 0.f);
}

extern "C" void kernel_launch(void* const* d_in, const int* in_sizes, int n_in,
                              void* d_out, int out_size, void* d_ws, size_t ws_size,
                              hipStream_t stream) {
  const int* ptr_t    = (const int*)d_in[0];
  const int* a_list_t = (const int*)d_in[1];
  const int* v_list_t = (const int*)d_in[2];
  const int* ptr_v    = (const int*)d_in[3];
  const int* a_list_v = (const int*)d_in[4];
  const int* t_list_v = (const int*)d_in[5];
  const float* t_embed = (const float*)d_in[6];
  const float* v_embed = (const float*)d_in[7];
  const float* a_embed = (const float*)d_in[8];
  const float* a_recv  = (const float*)d_in[9];
  const float* v_recv  = (const float*)d_in[10];
  const float* wv   = (const float*)d_in[11];
  const float* wt   = (const float*)d_in[12];
  const float* wa_v = (const float*)d_in[13];
  const float* wa_t = (const float*)d_in[14];
  const float* w1   = (const float*)d_in[15];
  const float* w2   = (const float*)d_in[16];
  const float* wa   = (const float*)d_in[17];

  const int NT = in_sizes[0] - 1;            // 50000
  const int NV = in_sizes[3] - 1;            // 50000
  const int NA = in_sizes[8] / DIM;          // 50000
  const int E  = in_sizes[1];                // 400000

  float* ws_t = (float*)d_ws;                // (NT,128) segment sums
  float* ws_v = ws_t + (size_t)NT * DIM;     // (NV,128) segment sums

  const int n4 = (NT + NV) * DIM / 4;
  zero_ws<<<(n4 + 255) / 256, 256, 0, stream>>>((float4*)d_ws, n4);

  const size_t smem_dual   = (size_t)(4 * 16 * LDR) * sizeof(float) + 16 * sizeof(int);
  const size_t smem_single = (size_t)(2 * 16 * LDR) * sizeof(float) + 16 * sizeof(int);

  // t-path: gathered half + recv half fused, *0.5 into ws_t
  edge_agg_wmma<true><<<E / 16, 256, smem_dual, stream>>>(
      a_embed, a_list_t, v_embed, v_list_t, a_recv, v_recv,
      wa_v, wv, ptr_t, NT, 0.5f, ws_t);
  // v-path into ws_v
  edge_agg_wmma<false><<<E / 16, 256, smem_single, stream>>>(
      a_embed, a_list_v, t_embed, t_list_v, nullptr, nullptr,
      wa_t, wt, ptr_v, NV, 1.0f, ws_v);

  float* out = (float*)d_out;
  final_concat_gemm<<<NT / 16, 256, 0, stream>>>(t_embed, ws_t, w1, out);
  final_concat_gemm<<<NV / 16, 256, 0, stream>>>(v_embed, ws_v, w2,
                                                 out + (size_t)NT * DIM);
  final_gemm_nn<<<NA / 16, 256, 0, stream>>>(a_embed, wa,
                                             out + (size_t)(NT + NV) * DIM);
}